// MGCN_395136991408
// MI455X (gfx1250) — compile-verified
//
#include <hip/hip_runtime.h>
#include <hip/hip_bf16.h>
#include <math.h>

// Problem constants (from reference)
constexpr int kB  = 128;   // batch
constexpr int kN  = 100;   // nodes
constexpr int kC  = 32;    // input channels
constexpr int kHE = 32;    // edge-MLP hidden
constexpr int kNP = 112;   // nodes padded to 7*16 for WMMA tiles
constexpr int kFO = 64;    // GCN layer width
// LDS stride for column-major g: even (8B-aligned b64 reads) and 54c%64 injective
// over c=0..15 but 54*dc==2 (mod 64) holds at dc=-13 -> lanes (c,hs=0),(c+13,hs=1) share banks.
constexpr int kGS = 118;

typedef float v2f __attribute__((ext_vector_type(2)));
typedef float v8f __attribute__((ext_vector_type(8)));

// D = A(16x4, f32) * B(4x16, f32) + C(16x16, f32), full-precision fp32 matrix pipe.
__device__ __forceinline__ v8f wmma4(v2f a, v2f b, v8f c) {
  return __builtin_amdgcn_wmma_f32_16x16x4_f32(false, a, false, b, (short)0, c,
                                               false, false);
}

// ---------------------------------------------------------------------------
// K1: hij[B*N, 64] = x[B*N, 32] @ [ew1[0:32,:] | ew1[32:64,:]]
// grid 400 blocks x 256 threads; 8 waves = 2 M-tiles x 4 N-tiles per block.
// ---------------------------------------------------------------------------
__global__ __launch_bounds__(256) void k_hij(const float* __restrict__ x,
                                             const float* __restrict__ ew1,
                                             float* __restrict__ hij) {
  const int tid  = threadIdx.x;
  const int wave = tid >> 5, lane = tid & 31;
  const int lr = lane & 15, hs = lane >> 4;
  const int mt = blockIdx.x * 2 + (wave >> 2);
  const int nt = wave & 3;
  const int row = mt * 16 + lr;          // 0..12799
  const int e   = nt * 16 + lr;          // 0..63
  const int wbase = (e >= 32) ? 32 : 0;  // hi-half: ew1 rows 0..31, hj-half: rows 32..63
  const int ec = e & 31;
  v8f acc = {0.f, 0.f, 0.f, 0.f, 0.f, 0.f, 0.f, 0.f};
  for (int k = 0; k < kC; k += 4) {
    const int kk = k + hs * 2;
    v2f a, b;
    a.x = x[row * kC + kk];
    a.y = x[row * kC + kk + 1];
    b.x = ew1[(kk + wbase) * kHE + ec];
    b.y = ew1[(kk + 1 + wbase) * kHE + ec];
    acc = wmma4(a, b, acc);
  }
#pragma unroll
  for (int v = 0; v < 8; ++v) {
    const int r = mt * 16 + v + hs * 8;
    hij[r * 64 + e] = acc[v];
  }
}

// ---------------------------------------------------------------------------
// K2: per-batch edge MLP -> A_pred (padded [NP,NP], zero diag/padding).
// Nonlinear per-(i,j,k) ReLU => VALU (only ~0.08 GFLOP total).
// ---------------------------------------------------------------------------
__global__ __launch_bounds__(256) void k_edge(const float* __restrict__ hij,
                                              const float* __restrict__ mask,
                                              const float* __restrict__ eb1,
                                              const float* __restrict__ ew2,
                                              const float* __restrict__ eb2,
                                              float* __restrict__ apred) {
  __shared__ float s_hi[kN][kHE];
  __shared__ float s_hj[kN][kHE];
  __shared__ float s_m[kN];
  __shared__ float s_eb1[kHE];
  __shared__ float s_ew2[kHE];
  const int b = blockIdx.x, tid = threadIdx.x;
  for (int idx = tid; idx < kN * 64; idx += blockDim.x) {
    const int n = idx >> 6, c = idx & 63;
    const float v = hij[(b * kN + n) * 64 + c];
    if (c < 32) s_hi[n][c] = v; else s_hj[n][c - 32] = v;
  }
  for (int n = tid; n < kN; n += blockDim.x) s_m[n] = mask[b * kN + n];
  if (tid < kHE) { s_eb1[tid] = eb1[tid]; s_ew2[tid] = ew2[tid]; }
  __syncthreads();
  const float eb2v = eb2[0];
  float* ap = apred + (size_t)b * kNP * kNP;
  // zero padding rows/cols and diagonal (disjoint from pair writes below)
  for (int idx = tid; idx < kNP * kNP; idx += blockDim.x) {
    const int i = idx / kNP, j = idx % kNP;
    if (i >= kN || j >= kN || i == j) ap[idx] = 0.f;
  }
  for (int p = tid; p < kN * kN; p += blockDim.x) {
    const int i = p / kN, j = p % kN;
    if (j <= i) continue;
    float sij = 0.f, sji = 0.f;
#pragma unroll 8
    for (int k = 0; k < kHE; ++k) {
      const float a  = s_hi[i][k] + s_hj[j][k] + s_eb1[k];
      const float bb = s_hi[j][k] + s_hj[i][k] + s_eb1[k];
      sij += fmaxf(a, 0.f) * s_ew2[k];
      sji += fmaxf(bb, 0.f) * s_ew2[k];
    }
    const float y = expf(0.5f * (sij + sji) + eb2v) * s_m[i] * s_m[j];
    ap[i * kNP + j] = y;
    ap[j * kNP + i] = y;
  }
}

// ---------------------------------------------------------------------------
// K3: L = D^-1/2 (A + I) D^-1/2. blockIdx.y: 0 -> from input A into L0,
// 1 -> in-place on A_pred buffer (L1). Matrices symmetric => row sums == col sums.
// ---------------------------------------------------------------------------
__global__ __launch_bounds__(256) void k_lap(const float* __restrict__ A,
                                             float* __restrict__ L0,
                                             float* __restrict__ L1) {
  __shared__ float s_dinv[kN];
  const int b = blockIdx.x, r = blockIdx.y, tid = threadIdx.x;
  const float* srcA = A + (size_t)b * kN * kN;
  float* buf1 = L1 + (size_t)b * kNP * kNP;
  float* dst  = (r == 0) ? (L0 + (size_t)b * kNP * kNP) : buf1;
  for (int i = tid; i < kN; i += blockDim.x) {
    float s = 0.f;
    if (r == 0) {
      for (int j = 0; j < kN; ++j) s += srcA[i * kN + j];
    } else {
      for (int j = 0; j < kN; ++j) s += buf1[i * kNP + j];
    }
    s += 1.f;  // + I
    s_dinv[i] = rsqrtf(s + 1e-5f);
  }
  __syncthreads();
  for (int idx = tid; idx < kNP * kNP; idx += blockDim.x) {
    const int i = idx / kNP, j = idx % kNP;
    float out = 0.f;
    if (i < kN && j < kN) {
      float a = (r == 0) ? srcA[i * kN + j] : buf1[i * kNP + j];
      if (i == j) a += 1.f;
      out = s_dinv[i] * a * s_dinv[j];
    }
    dst[idx] = out;
  }
}

// ---------------------------------------------------------------------------
// K4: copy x -> h0 padded to [NP, 32] with zero rows 100..111.
// ---------------------------------------------------------------------------
__global__ __launch_bounds__(256) void k_pad(const float* __restrict__ x,
                                             float* __restrict__ h0) {
  const int b = blockIdx.x, tid = threadIdx.x;
  for (int idx = tid; idx < kNP * kC; idx += blockDim.x) {
    const int n = idx / kC, c = idx % kC;
    h0[(size_t)b * kNP * kC + idx] = (n < kN) ? x[(b * kN + n) * kC + c] : 0.f;
  }
}

// ---------------------------------------------------------------------------
// K5: one GCN layer:  h_out = relu((L0@(h@Wa) + L1@(h@Wb) + bias) * mask)
// Phase 1: g[rel] = h @ W[rel]   (WMMA; column-major in LDS, stride kGS)
// Phase 2: acc = sum_rel L[rel] @ g[rel]; fused bias/mask/relu on store.
// One block per batch. A wave owns a full 16-row strip (4 N-tiles, 4 accums):
// one b64 A-load feeds 4 WMMAs; B-operand is one ds_load_b64.
// ---------------------------------------------------------------------------
__global__ __launch_bounds__(256) void k_layer(const float* __restrict__ h_in,
                                               int Fin,
                                               const float* __restrict__ W,
                                               const float* __restrict__ bias,
                                               const float* __restrict__ L0,
                                               const float* __restrict__ L1,
                                               const float* __restrict__ mask,
                                               float* __restrict__ h_out) {
  __shared__ float g[2][kFO][kGS];  // 2*64*118*4 = 60.4 KB, column-major
  __shared__ float s_m[kNP];
  const int b = blockIdx.x, tid = threadIdx.x;
  const int wave = tid >> 5, lane = tid & 31, lr = lane & 15, hs = lane >> 4;
  const int ko = hs * 2;
  const float* hb = h_in + (size_t)b * kNP * Fin;

  for (int i = tid; i < kNP; i += blockDim.x)
    s_m[i] = (i < kN) ? mask[b * kN + i] : 0.f;

  // Warm WGP$ with the L rows this wave will stream in phase 2
  // (__builtin_prefetch -> global_prefetch_b8).
  if (wave < 7) {
    const int arow = wave * 16 + lr;
    const char* p0 = (const char*)(L0 + (size_t)b * kNP * kNP + arow * kNP);
    const char* p1 = (const char*)(L1 + (size_t)b * kNP * kNP + arow * kNP);
    __builtin_prefetch(p0 + hs * 128, 0, 0);
    __builtin_prefetch(p0 + 256 + hs * 128, 0, 0);
    __builtin_prefetch(p1 + hs * 128, 0, 0);
    __builtin_prefetch(p1 + 256 + hs * 128, 0, 0);
  }

  // Phase 1: g[rel] = h @ W_rel  ([NP,Fin] @ [Fin,64]); 14 row-strips / 8 waves.
  const v8f vzero = {0.f, 0.f, 0.f, 0.f, 0.f, 0.f, 0.f, 0.f};
  for (int gi = wave; gi < 14; gi += 8) {
    const int rel = gi / 7, mt = gi % 7;
    const float* Ws = W + rel * Fin * kFO;
    const int arow = mt * 16 + lr;
    v8f acc[4] = {vzero, vzero, vzero, vzero};
    for (int k = 0; k < Fin; k += 4) {
      const int kk = k + ko;
      v2f a;
      a.x = hb[arow * Fin + kk];
      a.y = hb[arow * Fin + kk + 1];
#pragma unroll
      for (int nt = 0; nt < 4; ++nt) {
        v2f bb;
        bb.x = Ws[kk * kFO + nt * 16 + lr];
        bb.y = Ws[(kk + 1) * kFO + nt * 16 + lr];
        acc[nt] = wmma4(a, bb, acc[nt]);
      }
    }
#pragma unroll
    for (int nt = 0; nt < 4; ++nt)
#pragma unroll
      for (int v = 0; v < 8; ++v)
        g[rel][nt * 16 + lr][mt * 16 + v + hs * 8] = acc[nt][v];
  }
  __syncthreads();

  // Phase 2: acc = L0@g0 + L1@g1; one b64 L-load -> 4 WMMAs.
  for (int mt = wave; mt < 7; mt += 8) {
    const int arow = mt * 16 + lr;
    v8f acc[4] = {vzero, vzero, vzero, vzero};
#pragma unroll
    for (int rel = 0; rel < 2; ++rel) {
      const float* Lrow = (rel ? L1 : L0) + (size_t)b * kNP * kNP + arow * kNP;
      for (int k = 0; k < kNP; k += 4) {
        const int kk = k + ko;
        v2f a;
        a.x = Lrow[kk];
        a.y = Lrow[kk + 1];
#pragma unroll
        for (int nt = 0; nt < 4; ++nt) {
          const v2f bb = *(const v2f*)&g[rel][nt * 16 + lr][kk];  // ds_load_b64
          acc[nt] = wmma4(a, bb, acc[nt]);
        }
      }
    }
#pragma unroll
    for (int nt = 0; nt < 4; ++nt) {
      const int bcol = nt * 16 + lr;
      const float bc = bias[bcol];
#pragma unroll
      for (int v = 0; v < 8; ++v) {
        const int row = mt * 16 + v + hs * 8;
        const float val = fmaxf((acc[nt][v] + bc) * s_m[row], 0.f);
        h_out[(size_t)b * kNP * kFO + row * kFO + bcol] = val;  // padded rows -> 0
      }
    }
  }
}

// ---------------------------------------------------------------------------
// K6: max-pool over nodes + final [64,16] linear.
// ---------------------------------------------------------------------------
__global__ __launch_bounds__(64) void k_final(const float* __restrict__ h,
                                              const float* __restrict__ fw,
                                              const float* __restrict__ fb,
                                              float* __restrict__ out) {
  __shared__ float s_p[kFO];
  const int b = blockIdx.x, f = threadIdx.x;
  float m = -INFINITY;
  for (int n = 0; n < kN; ++n)
    m = fmaxf(m, h[(size_t)b * kNP * kFO + n * kFO + f]);
  s_p[f] = m;
  __syncthreads();
  if (f < 16) {
    float acc = fb[f];
    for (int k = 0; k < kFO; ++k) acc += s_p[k] * fw[k * 16 + f];
    out[b * 16 + f] = acc;
  }
}

// ---------------------------------------------------------------------------
extern "C" void kernel_launch(void* const* d_in, const int* in_sizes, int n_in,
                              void* d_out, int out_size, void* d_ws, size_t ws_size,
                              hipStream_t stream) {
  const float* x    = (const float*)d_in[0];
  const float* A    = (const float*)d_in[1];
  const float* mask = (const float*)d_in[2];
  const float* ew1  = (const float*)d_in[3];
  const float* eb1  = (const float*)d_in[4];
  const float* ew2  = (const float*)d_in[5];
  const float* eb2  = (const float*)d_in[6];
  const float* gw0  = (const float*)d_in[7];
  const float* gb0  = (const float*)d_in[8];
  const float* gw1  = (const float*)d_in[9];
  const float* gb1  = (const float*)d_in[10];
  const float* gw2  = (const float*)d_in[11];
  const float* gb2  = (const float*)d_in[12];
  const float* fw   = (const float*)d_in[13];
  const float* fb   = (const float*)d_in[14];

  // workspace layout (floats), total ~23.5 MB
  float* ws  = (float*)d_ws;
  float* hij = ws;                                // B*N*64
  float* L0  = hij + (size_t)kB * kN * 64;        // B*NP*NP
  float* L1  = L0 + (size_t)kB * kNP * kNP;       // B*NP*NP (A_pred, then in-place L1)
  float* hA  = L1 + (size_t)kB * kNP * kNP;       // B*NP*64
  float* hB  = hA + (size_t)kB * kNP * 64;        // B*NP*64

  k_hij<<<400, 256, 0, stream>>>(x, ew1, hij);
  k_edge<<<kB, 256, 0, stream>>>(hij, mask, eb1, ew2, eb2, L1);
  k_lap<<<dim3(kB, 2), 256, 0, stream>>>(A, L0, L1);
  k_pad<<<kB, 256, 0, stream>>>(x, hA);
  k_layer<<<kB, 256, 0, stream>>>(hA, kC, gw0, gb0, L0, L1, mask, hB);
  k_layer<<<kB, 256, 0, stream>>>(hB, 64, gw1, gb1, L0, L1, mask, hA);
  k_layer<<<kB, 256, 0, stream>>>(hA, 64, gw2, gb2, L0, L1, mask, hB);
  k_final<<<kB, 64, 0, stream>>>(hB, fw, fb, (float*)d_out);
}